// ResnetNetwork_43550968382120
// MI455X (gfx1250) — compile-verified
//
#include <hip/hip_runtime.h>
#include <hip/hip_bf16.h>
#include <math.h>

typedef float v2f __attribute__((ext_vector_type(2)));
typedef float v8f __attribute__((ext_vector_type(8)));

namespace {
constexpr int kD    = 2048;  // D_FEAT
constexpr int kC    = 9;     // N_CLASSES
constexpr int kInst = 64;    // instances per bag
constexpr int kBags = 256;   // batch
}

// One block per bag. 4 waves, each wave owns a 16-row tile of x.
// Computes per-bag sum of weights * BCEWithLogits(max-pooled logits, y).
__global__ __launch_bounds__(128) void mil_bag_wmma_kernel(
    const float* __restrict__ x,     // [256, 64, 2048]
    const float* __restrict__ y,     // [256, 9]
    const float* __restrict__ wts,   // [256, 9]
    const float* __restrict__ W,     // [9, 2048]
    const float* __restrict__ bias,  // [9]
    float* __restrict__ partial)     // [256]
{
  const int bag  = blockIdx.x;
  const int wave = threadIdx.x >> 5;   // 0..3 -> row tile within the bag
  const int lane = threadIdx.x & 31;
  const int half = lane >> 4;          // 0: K=k,k+1   1: K=k+2,k+3
  const int l    = lane & 15;          // A: row M ; B: column N

  // A-matrix source row (instance) for this lane
  const float* xrow = x + ((size_t)bag * kInst + (size_t)wave * 16 + l) * kD;
  // B-matrix source: column n=l takes W row l (garbage for l>=9, ignored later)
  const float* wrow = W + (size_t)((l < kC) ? l : 0) * kD;
  const int koff = half * 2;

  v8f acc = {0.f, 0.f, 0.f, 0.f, 0.f, 0.f, 0.f, 0.f};

#pragma unroll 8
  for (int k = 0; k < kD; k += 4) {
    // A 16x4 f32 fragment: lane l holds x[row][k+koff], x[row][k+koff+1]
    v2f a = *reinterpret_cast<const v2f*>(xrow + k + koff);
    // B 4x16 f32 fragment: lane l holds W[n][k+koff], W[n][k+koff+1]
    v2f b = *reinterpret_cast<const v2f*>(wrow + k + koff);
    // 8-arg form: (neg_a, A, neg_b, B, c_mod, C, reuse_a, reuse_b)
    acc = __builtin_amdgcn_wmma_f32_16x16x4_f32(
        false, a, false, b, (short)0, acc, false, false);
  }

  // Max over this tile's 16 rows.
  // acc[m]: lanes 0-15 -> row m, lanes 16-31 -> row m+8 (column n = l).
  float r = acc[0];
  r = fmaxf(r, acc[1]); r = fmaxf(r, acc[2]); r = fmaxf(r, acc[3]);
  r = fmaxf(r, acc[4]); r = fmaxf(r, acc[5]); r = fmaxf(r, acc[6]);
  r = fmaxf(r, acc[7]);
  // combine the two row-halves (lane <-> lane^16)
  r = fmaxf(r, __shfl_xor(r, 16, 32));
  // fold the reference's x*2 scale (monotone, commutes with max)
  r *= 2.0f;

  __shared__ float smax[4][16];
  __shared__ float contrib[16];
  if (half == 0) smax[wave][l] = r;
  __syncthreads();

  if (threadIdx.x < kC) {
    const int n = threadIdx.x;
    float z = fmaxf(fmaxf(smax[0][n], smax[1][n]),
                    fmaxf(smax[2][n], smax[3][n]));
    z += bias[n];  // bias commutes with the instance max
    const float t = y[bag * kC + n];
    const float w = wts[bag * kC + n];
    // stable BCE-with-logits: max(z,0) - z*t + log1p(exp(-|z|))
    const float bce = fmaxf(z, 0.f) - z * t + log1pf(expf(-fabsf(z)));
    contrib[n] = w * bce;
  }
  __syncthreads();

  if (threadIdx.x == 0) {
    float s = 0.f;
#pragma unroll
    for (int n = 0; n < kC; ++n) s += contrib[n];
    partial[bag] = s;
  }
}

// Deterministic tree reduction of 256 per-bag partials -> mean over 256*9.
__global__ __launch_bounds__(256) void mil_final_reduce_kernel(
    const float* __restrict__ partial, float* __restrict__ out)
{
  __shared__ float s[256];
  const int t = threadIdx.x;
  s[t] = partial[t];
  __syncthreads();
  for (int off = 128; off > 0; off >>= 1) {
    if (t < off) s[t] += s[t + off];
    __syncthreads();
  }
  if (t == 0) out[0] = s[0] / (float)(kBags * kC);
}

extern "C" void kernel_launch(void* const* d_in, const int* in_sizes, int n_in,
                              void* d_out, int out_size, void* d_ws, size_t ws_size,
                              hipStream_t stream) {
  const float* x    = (const float*)d_in[0];  // [256,64,2048]
  const float* y    = (const float*)d_in[1];  // [256,9]
  const float* wts  = (const float*)d_in[2];  // [256,9]
  const float* W    = (const float*)d_in[3];  // [9,2048]
  const float* bias = (const float*)d_in[4];  // [9]
  float* partial = (float*)d_ws;              // 256 floats of scratch
  float* out     = (float*)d_out;

  mil_bag_wmma_kernel<<<kBags, 128, 0, stream>>>(x, y, wts, W, bias, partial);
  mil_final_reduce_kernel<<<1, 256, 0, stream>>>(partial, out);
}